// MergedGCN_69423851373101
// MI455X (gfx1250) — compile-verified
//
#include <hip/hip_runtime.h>

#define NNODES 50000
#define DIN    64
#define DH     128
#define EPSV   1e-5f
#define SLOPE  0.1f
#define RT     5                    // row tiles (16 rows each) per GEMM block
#define PAD0   (DIN + 4)            // padded LDS row stride (floats), 16B-aligned
#define PAD1   (DH + 4)

typedef __attribute__((ext_vector_type(2))) float v2f;
typedef __attribute__((ext_vector_type(8))) float v8f;
typedef int v4i __attribute__((vector_size(16)));

// ---- async global->LDS copy of 16 bytes (CDNA5 GLOBAL_LOAD_ASYNC_TO_LDS_B128)
__device__ __forceinline__ void async_cp16(const float* g, float* l) {
#if __has_builtin(__builtin_amdgcn_global_load_async_to_lds_b128)
    __builtin_amdgcn_global_load_async_to_lds_b128(
        (__attribute__((address_space(1))) v4i*)g,
        (__attribute__((address_space(3))) v4i*)l,
        0, 0);
#else
    *(float4*)l = *(const float4*)g;
#endif
}
__device__ __forceinline__ void async_wait0() {
#if __has_builtin(__builtin_amdgcn_s_wait_asynccnt)
    __builtin_amdgcn_s_wait_asynccnt(0);
#elif __has_builtin(__builtin_amdgcn_global_load_async_to_lds_b128)
    asm volatile("s_wait_asynccnt 0" ::: "memory");
#endif
}

// ---------------- degree / norm ----------------
__global__ void k_deg_init(float* deg) {
    int i = blockIdx.x * blockDim.x + threadIdx.x;
    if (i < NNODES) deg[i] = 1.0f;              // self-loop weight
}
__global__ void k_deg_acc(const int* __restrict__ dst, const float* __restrict__ ew,
                          float* __restrict__ deg, int nE) {
    int i = blockIdx.x * blockDim.x + threadIdx.x;
    if (i < nE) atomicAdd(&deg[dst[i]], ew[i]);
}
__global__ void k_deg_inv(float* deg) {
    int i = blockIdx.x * blockDim.x + threadIdx.x;
    if (i < NNODES) deg[i] = rsqrtf(deg[i]);    // deg >= 1 always
}

// -------- fused dual GEMM (layer 0): H = X@W0, R = X@Wres + bres ----------
// B panels live in registers for the whole block; A tiles staged in LDS via
// async copies and shared by all 8 waves. 5 row tiles per block.
__global__ void k_gemm0_dual(const float* __restrict__ X,
                             const float* __restrict__ W0,
                             const float* __restrict__ Wr,
                             const float* __restrict__ br,
                             float* __restrict__ H,
                             float* __restrict__ R) {
    __shared__ float shA[16 * PAD0];            // 16 x 64 tile, padded rows
    const int wave = threadIdx.x >> 5;          // 0..7 -> column tile
    const int lane = threadIdx.x & 31;
    const int hi   = lane >> 4;                 // K-pair select
    const int l15  = lane & 15;
    const int colBase = wave * 16;

    // preload both B panels (K=64 -> 16 v2f each) into registers
    const float* bp0 = W0 + (size_t)(2 * hi) * DH + colBase + l15;
    const float* bpr = Wr + (size_t)(2 * hi) * DH + colBase + l15;
    v2f b0r[DIN / 4], brr[DIN / 4];
#pragma unroll
    for (int kk = 0; kk < DIN / 4; ++kk) {
        b0r[kk].x = bp0[(size_t)(4 * kk) * DH];
        b0r[kk].y = bp0[(size_t)(4 * kk + 1) * DH];
        brr[kk].x = bpr[(size_t)(4 * kk) * DH];
        brr[kk].y = bpr[(size_t)(4 * kk + 1) * DH];
    }
    const float biasr = br[colBase + l15];

    // copy mapping: 256 threads x 1 float4 covers 16x64 tile
    const int crow = threadIdx.x >> 4;          // 0..15
    const int coff = (threadIdx.x & 15) * 4;    // 0..60

    for (int rt = 0; rt < RT; ++rt) {
        const int rowBase = blockIdx.x * (16 * RT) + rt * 16;
        if (rt) __syncthreads();                // prior tile fully consumed
        async_cp16(X + (size_t)(rowBase + crow) * DIN + coff,
                   shA + crow * PAD0 + coff);
        async_wait0();
        __syncthreads();                        // tile visible to all waves

        const float* a_lds = shA + l15 * PAD0 + 2 * hi;
        v8f acc0 = {}; v8f accr = {};
#pragma unroll
        for (int k = 0; k < DIN; k += 4) {
            v2f a = *(const v2f*)(a_lds + k);
            acc0 = __builtin_amdgcn_wmma_f32_16x16x4_f32(false, a, false, b0r[k >> 2],
                                                         (short)0, acc0, false, false);
            accr = __builtin_amdgcn_wmma_f32_16x16x4_f32(false, a, false, brr[k >> 2],
                                                         (short)0, accr, false, false);
        }
#pragma unroll
        for (int r = 0; r < 8; ++r) {
            int row = rowBase + r + 8 * hi;
            size_t idx = (size_t)row * DH + colBase + l15;
            H[idx] = acc0[r];                   // conv bias added post-aggregation
            R[idx] = accr[r] + biasr;
        }
    }
}

// ---------------- GEMM (layer 1): H = A@W1, K = DH ----------------
__global__ void k_gemm1(const float* __restrict__ A,
                        const float* __restrict__ W,
                        float* __restrict__ H) {
    __shared__ float shA[16 * PAD1];            // 16 x 128 tile, padded rows
    const int wave = threadIdx.x >> 5;
    const int lane = threadIdx.x & 31;
    const int hi   = lane >> 4;
    const int l15  = lane & 15;
    const int colBase = wave * 16;

    // preload B panel (K=128 -> 32 v2f) into registers
    const float* bp = W + (size_t)(2 * hi) * DH + colBase + l15;
    v2f breg[DH / 4];
#pragma unroll
    for (int kk = 0; kk < DH / 4; ++kk) {
        breg[kk].x = bp[(size_t)(4 * kk) * DH];
        breg[kk].y = bp[(size_t)(4 * kk + 1) * DH];
    }

    // copy mapping: 512 float4 chunks, 2 per thread
    const int c0row = threadIdx.x >> 5;         // chunks 0..255: rows 0..7
    const int c0off = (threadIdx.x & 31) * 4;
    const int c1row = c0row + 8;                // chunks 256..511: rows 8..15

    for (int rt = 0; rt < RT; ++rt) {
        const int rowBase = blockIdx.x * (16 * RT) + rt * 16;
        if (rt) __syncthreads();
        async_cp16(A + (size_t)(rowBase + c0row) * DH + c0off,
                   shA + c0row * PAD1 + c0off);
        async_cp16(A + (size_t)(rowBase + c1row) * DH + c0off,
                   shA + c1row * PAD1 + c0off);
        async_wait0();
        __syncthreads();

        const float* a_lds = shA + l15 * PAD1 + 2 * hi;
        v8f acc = {};
#pragma unroll
        for (int k = 0; k < DH; k += 4) {
            v2f a = *(const v2f*)(a_lds + k);
            acc = __builtin_amdgcn_wmma_f32_16x16x4_f32(false, a, false, breg[k >> 2],
                                                        (short)0, acc, false, false);
        }
#pragma unroll
        for (int r = 0; r < 8; ++r) {
            int row = rowBase + r + 8 * hi;
            H[(size_t)row * DH + colBase + l15] = acc[r];
        }
    }
}

// ---------------- aggregation: self-loop + bias init ----------------
__global__ void k_agg_init(const float* __restrict__ h, float* __restrict__ agg,
                           const float* __restrict__ dinv, const float* __restrict__ bias) {
    int i = blockIdx.x * blockDim.x + threadIdx.x;
    if (i < NNODES * DH) {
        int c = i & (DH - 1);
        int n = i >> 7;
        float di = dinv[n];
        agg[i] = bias[c] + di * di * h[i];
    }
}

// ---------------- aggregation: one wave per edge, float4 per lane ----------
__global__ void k_agg_edges(const float* __restrict__ h, float* __restrict__ agg,
                            const int* __restrict__ src, const int* __restrict__ dst,
                            const float* __restrict__ ew, const float* __restrict__ dinv,
                            int nE) {
    int gid  = blockIdx.x * blockDim.x + threadIdx.x;
    int e    = gid >> 5;
    int lane = gid & 31;
    if (e >= nE) return;
    int s = src[e], d = dst[e];
    float nrm = dinv[s] * ew[e] * dinv[d];
    float4 v = ((const float4*)(h + (size_t)s * DH))[lane];
    float* ad = agg + (size_t)d * DH + lane * 4;
    atomicAdd(ad + 0, nrm * v.x);
    atomicAdd(ad + 1, nrm * v.y);
    atomicAdd(ad + 2, nrm * v.z);
    atomicAdd(ad + 3, nrm * v.w);
}

// ---------------- BN stats (column mean / mean-square) ----------------
__global__ void k_zero_stats(float* stats) {
    int i = blockIdx.x * blockDim.x + threadIdx.x;
    if (i < 2 * DH) stats[i] = 0.0f;
}
__global__ void k_bn_stats(const float* __restrict__ h, float* __restrict__ stats) {
    int c = threadIdx.x;                        // blockDim = DH
    float s = 0.0f, q = 0.0f;
    for (int r = blockIdx.x; r < NNODES; r += gridDim.x) {
        float v = h[(size_t)r * DH + c];
        s += v; q += v * v;
    }
    atomicAdd(&stats[c], s);
    atomicAdd(&stats[DH + c], q);
}

// ---------------- BN apply + leaky relu + residual ----------------
__global__ void k_bn_apply(const float* __restrict__ agg, const float* __restrict__ stats,
                           const float* __restrict__ g, const float* __restrict__ be,
                           const float* __restrict__ res, float* __restrict__ out) {
    int i = blockIdx.x * blockDim.x + threadIdx.x;
    if (i < NNODES * DH) {
        int c = i & (DH - 1);
        const float invN = 1.0f / (float)NNODES;
        float mu  = stats[c] * invN;
        float var = stats[DH + c] * invN - mu * mu;
        float v = (agg[i] - mu) * rsqrtf(var + EPSV) * g[c] + be[c];
        v = (v >= 0.0f) ? v : SLOPE * v;
        out[i] = v + res[i];
    }
}

// ---------------- final layer norm (one wave per row) ----------------
__global__ void k_layer_norm(const float* __restrict__ h, float* __restrict__ out) {
    int gid  = blockIdx.x * blockDim.x + threadIdx.x;
    int row  = gid >> 5;
    int lane = gid & 31;
    if (row >= NNODES) return;
    float4 v = ((const float4*)(h + (size_t)row * DH))[lane];
    float s = v.x + v.y + v.z + v.w;
#pragma unroll
    for (int o = 16; o > 0; o >>= 1) s += __shfl_xor(s, o, 32);
    float mu = s * (1.0f / DH);
    float dx = v.x - mu, dy = v.y - mu, dz = v.z - mu, dw = v.w - mu;
    float q = dx * dx + dy * dy + dz * dz + dw * dw;
#pragma unroll
    for (int o = 16; o > 0; o >>= 1) q += __shfl_xor(q, o, 32);
    float inv = rsqrtf(q * (1.0f / DH) + EPSV);
    float4 r; r.x = dx * inv; r.y = dy * inv; r.z = dz * inv; r.w = dw * inv;
    ((float4*)(out + (size_t)row * DH))[lane] = r;
}

extern "C" void kernel_launch(void* const* d_in, const int* in_sizes, int n_in,
                              void* d_out, int out_size, void* d_ws, size_t ws_size,
                              hipStream_t stream) {
    const float* x    = (const float*)d_in[0];
    const int*   src  = (const int*)  d_in[1];
    const int*   dst  = (const int*)  d_in[2];
    const float* ew   = (const float*)d_in[3];
    const float* W0   = (const float*)d_in[4];
    const float* b0   = (const float*)d_in[5];
    const float* g0   = (const float*)d_in[6];
    const float* be0  = (const float*)d_in[7];
    const float* W1   = (const float*)d_in[8];
    const float* b1   = (const float*)d_in[9];
    const float* g1   = (const float*)d_in[10];
    const float* be1  = (const float*)d_in[11];
    const float* Wres = (const float*)d_in[12];
    const float* bres = (const float*)d_in[13];
    const int E = in_sizes[1];

    float* w     = (float*)d_ws;
    float* dinv  = w;                                   // N
    float* hbuf  = dinv + NNODES;                       // N*DH
    float* agg   = hbuf + (size_t)NNODES * DH;          // N*DH
    float* res   = agg  + (size_t)NNODES * DH;          // N*DH
    float* stats = res  + (size_t)NNODES * DH;          // 2*DH
    float* act   = (float*)d_out;                       // N*DH scratch / final out

    const int TB = 256;
    const int nElem   = NNODES * DH;
    const int gElem   = (nElem + TB - 1) / TB;
    const int gNodes  = (NNODES + TB - 1) / TB;
    const int gEdges  = (E + TB - 1) / TB;
    const int gEdgeWv = ((E * 32) + TB - 1) / TB;       // one wave per edge
    const int gGemm   = NNODES / (16 * RT);             // 625 blocks x 80 rows
    const int gLN     = ((NNODES * 32) + TB - 1) / TB;

    // normalization coefficients
    k_deg_init<<<gNodes, TB, 0, stream>>>(dinv);
    k_deg_acc <<<gEdges, TB, 0, stream>>>(dst, ew, dinv, E);
    k_deg_inv <<<gNodes, TB, 0, stream>>>(dinv);

    // ---- layer 0 ----
    k_gemm0_dual<<<gGemm, TB, 0, stream>>>(x, W0, Wres, bres, hbuf, res);
    k_agg_init  <<<gElem, TB, 0, stream>>>(hbuf, agg, dinv, b0);
    k_agg_edges <<<gEdgeWv, TB, 0, stream>>>(hbuf, agg, src, dst, ew, dinv, E);
    k_zero_stats<<<1, TB, 0, stream>>>(stats);
    k_bn_stats  <<<512, DH, 0, stream>>>(agg, stats);
    k_bn_apply  <<<gElem, TB, 0, stream>>>(agg, stats, g0, be0, res, act);

    // ---- layer 1 ----
    k_gemm1     <<<gGemm, TB, 0, stream>>>(act, W1, hbuf);
    k_agg_init  <<<gElem, TB, 0, stream>>>(hbuf, agg, dinv, b1);
    k_agg_edges <<<gEdgeWv, TB, 0, stream>>>(hbuf, agg, src, dst, ew, dinv, E);
    k_zero_stats<<<1, TB, 0, stream>>>(stats);
    k_bn_stats  <<<512, DH, 0, stream>>>(agg, stats);
    k_bn_apply  <<<gElem, TB, 0, stream>>>(agg, stats, g1, be1, res, hbuf);

    // ---- final layer norm ----
    k_layer_norm<<<gLN, TB, 0, stream>>>(hbuf, (float*)d_out);
}